// SpectralConv2d_70351564308906
// MI455X (gfx1250) — compile-verified
//
#include <hip/hip_runtime.h>
#include <hip/hip_bf16.h>

// ---------------------------------------------------------------------------
// SpectralConv2d (FNO) on gfx1250 via partial-DFT GEMMs with V_WMMA_F32_16X16X4_F32
// B=16, Ci=Co=64, H=W=256, modes: 32 rows (0..15, 240..255) x 16 cols
// ---------------------------------------------------------------------------

typedef float v2f __attribute__((ext_vector_type(2)));
typedef float v8f __attribute__((ext_vector_type(8)));

__device__ __forceinline__ v8f wmma_f32(v2f a, v2f b, v8f c) {
  // D = A(16x4 f32) x B(4x16 f32) + C(16x16 f32)
  return __builtin_amdgcn_wmma_f32_16x16x4_f32(
      false, a, false, b, (short)0, c, false, false);
}

// ---- workspace layout (float offsets) ----
// S1 : (B*Ci*H) x 32   forward-W DFT result [re cols 0..15 | im cols 16..31]
//      (reused later as G: (B*Co*H) x 32 inverse-H result)
// T  : per (b,c) 64x32 raw stage-2 GEMM output
// U2 : [mode(32*16)][b(16)][c(64)]  re/im  retained spectrum
// V  : [b(16)][r(32)][w'(16)][o(64)] re/im  after spectral multiply
// Wt : [mode][c][o] re/im transposed weights
// tables: Dw(256x32) Dh(64x256) Ere/Eim(256x64) Fw(32x256)
static constexpr size_t O_S1   = 0;               // 8388608
static constexpr size_t O_T    = 8388608;         // 2097152
static constexpr size_t O_U2RE = 10485760;        // 524288
static constexpr size_t O_U2IM = 11010048;        // 524288
static constexpr size_t O_VRE  = 11534336;        // 524288
static constexpr size_t O_VIM  = 12058624;        // 524288
static constexpr size_t O_WTRE = 12582912;        // 2097152
static constexpr size_t O_WTIM = 14680064;        // 2097152
static constexpr size_t O_DW   = 16777216;        // 8192
static constexpr size_t O_DH   = 16785408;        // 16384
static constexpr size_t O_ERE  = 16801792;        // 16384
static constexpr size_t O_EIM  = 16818176;        // 16384
static constexpr size_t O_FW   = 16834560;        // 8192

// ---------------------------------------------------------------------------
// Twiddle tables. 65536 threads exactly.
// ---------------------------------------------------------------------------
__global__ void k_prep_tables(float* __restrict__ ws) {
  int t = blockIdx.x * blockDim.x + threadIdx.x;
  const float s  = 1.0f / 16.0f;                    // 1/sqrt(256) per axis
  const float w0 = 6.2831853071795864769f / 256.0f; // 2*pi/256
  if (t < 8192) {                      // Dw[y][n]: forward W-DFT
    int y = t >> 5, n = t & 31;
    int wp = n & 15, comp = n >> 4;
    int ph = (y * wp) & 255;
    float th = w0 * (float)ph;
    ws[O_DW + t] = s * (comp == 0 ? __cosf(th) : -__sinf(th));
  } else if (t < 24576) {              // Dh[p][x]: forward H-DFT (cos rows | sin rows)
    int i = t - 8192;
    int p = i >> 8, x = i & 255;
    int r = p & 31;
    int hp = (r < 16) ? r : 224 + r;   // 240 + (r-16)
    int ph = (hp * x) & 255;
    float th = w0 * (float)ph;
    ws[O_DH + i] = s * ((p < 32) ? __cosf(th) : __sinf(th));
  } else if (t < 40960) {              // Ere[x][p]: inverse H, real part
    int i = t - 24576;
    int x = i >> 6, p = i & 63;
    int r = p & 31;
    int hp = (r < 16) ? r : 224 + r;
    int ph = (hp * x) & 255;
    float th = w0 * (float)ph;
    ws[O_ERE + i] = s * ((p < 32) ? __cosf(th) : -__sinf(th));
  } else if (t < 57344) {              // Eim[x][p]: inverse H, imag part
    int i = t - 40960;
    int x = i >> 6, p = i & 63;
    int r = p & 31;
    int hp = (r < 16) ? r : 224 + r;
    int ph = (hp * x) & 255;
    float th = w0 * (float)ph;
    ws[O_EIM + i] = s * ((p < 32) ? __sinf(th) : __cosf(th));
  } else {                             // Fw[n][y]: real irfft along W
    int i = t - 57344;
    int n = i >> 8, y = i & 255;
    float v;
    if (n == 0)        v = s;                                   // DC, Im dropped
    else if (n < 16) { int ph = (n * y) & 255;        v =  2.0f * s * __cosf(w0 * (float)ph); }
    else if (n == 16)  v = 0.0f;                                // Im of mode 0 dropped
    else             { int ph = ((n - 16) * y) & 255; v = -2.0f * s * __sinf(w0 * (float)ph); }
    ws[O_FW + i] = v;
  }
}

// ---------------------------------------------------------------------------
// Transpose weights (Ci,Co,16,16) -> [mode(32x16)][c][o], merging pos/neg.
// 2097152 threads.
// ---------------------------------------------------------------------------
__global__ void k_prep_weights(const float* __restrict__ Wpr, const float* __restrict__ Wpi,
                               const float* __restrict__ Wnr, const float* __restrict__ Wni,
                               float* __restrict__ ws) {
  int t = blockIdx.x * blockDim.x + threadIdx.x;
  int o = t & 63, c = (t >> 6) & 63, m = t >> 12;
  int r = m >> 4, wp = m & 15;
  int ri = (r < 16) ? r : (r - 16);
  long src = (((long)c * 64 + o) * 16 + ri) * 16 + wp;
  ws[O_WTRE + t] = (r < 16) ? Wpr[src] : Wnr[src];
  ws[O_WTIM + t] = (r < 16) ? Wpi[src] : Wni[src];
}

// ---------------------------------------------------------------------------
// Stage 1: S1(M1 x 32) = U(M1 x 256) x Dw(256 x 32).  16384 waves, 4 per block.
// ---------------------------------------------------------------------------
__global__ void k_dft_w(const float* __restrict__ u, float* __restrict__ ws) {
  const float* Dw = ws + O_DW;
  float* S1 = ws + O_S1;
  int gid  = blockIdx.x * blockDim.x + threadIdx.x;
  int wave = gid >> 5;
  int lane = threadIdx.x & 31;
  int ln = lane & 15, hi = lane >> 4;
  long rowbase = (long)wave * 16;
  const float* arow = u + (rowbase + ln) * 256;
  v8f acc0 = {}; v8f acc1 = {};
  for (int k0 = 0; k0 < 256; k0 += 4) {
    int k = k0 + hi * 2;
    v2f a = *(const v2f*)(arow + k);
    v2f b0, b1;
    b0.x = Dw[k * 32 + ln];       b0.y = Dw[(k + 1) * 32 + ln];
    b1.x = Dw[k * 32 + 16 + ln];  b1.y = Dw[(k + 1) * 32 + 16 + ln];
    acc0 = wmma_f32(a, b0, acc0);
    acc1 = wmma_f32(a, b1, acc1);
  }
  float* crow = S1 + rowbase * 32;
#pragma unroll
  for (int v = 0; v < 8; ++v) {
    int mr = v + 8 * hi;
    crow[(long)mr * 32 + ln]      = acc0[v];
    crow[(long)mr * 32 + 16 + ln] = acc1[v];
  }
}

// ---------------------------------------------------------------------------
// Stage 2: per (b,c): T(64x32) = Dh(64x256) x S1_bc(256x32). 4096 waves.
// ---------------------------------------------------------------------------
__global__ void k_dft_h(float* __restrict__ ws) {
  const float* Dh = ws + O_DH;
  const float* S1 = ws + O_S1;
  float* T = ws + O_T;
  int gid  = blockIdx.x * blockDim.x + threadIdx.x;
  int wave = gid >> 5;
  int lane = threadIdx.x & 31;
  int ln = lane & 15, hi = lane >> 4;
  int bc = wave >> 2, mt = wave & 3;
  const float* Bsrc = S1 + (long)bc * 256 * 32;
  v8f acc0 = {}; v8f acc1 = {};
  for (int k0 = 0; k0 < 256; k0 += 4) {
    int k = k0 + hi * 2;
    v2f a = *(const v2f*)(Dh + (mt * 16 + ln) * 256 + k);
    v2f b0, b1;
    b0.x = Bsrc[k * 32 + ln];       b0.y = Bsrc[(k + 1) * 32 + ln];
    b1.x = Bsrc[k * 32 + 16 + ln];  b1.y = Bsrc[(k + 1) * 32 + 16 + ln];
    acc0 = wmma_f32(a, b0, acc0);
    acc1 = wmma_f32(a, b1, acc1);
  }
  float* Trow = T + (long)bc * 2048 + mt * 16 * 32;
#pragma unroll
  for (int v = 0; v < 8; ++v) {
    int p = v + 8 * hi;
    Trow[p * 32 + ln]      = acc0[v];
    Trow[p * 32 + 16 + ln] = acc1[v];
  }
}

// ---------------------------------------------------------------------------
// Combine cos/sin partial sums into complex spectrum, transpose to
// U2[mode][b][c].  524288 threads.
// u_hat_re = T[r,w'] + T[32+r,16+w'] ;  u_hat_im = T[r,16+w'] - T[32+r,w']
// ---------------------------------------------------------------------------
__global__ void k_combine(float* __restrict__ ws) {
  int t = blockIdx.x * blockDim.x + threadIdx.x;
  int wp = t & 15, r = (t >> 4) & 31, c = (t >> 9) & 63, b = t >> 15;
  const float* Tb = ws + O_T + (long)(b * 64 + c) * 2048;
  float re = Tb[r * 32 + wp]      + Tb[(32 + r) * 32 + 16 + wp];
  float im = Tb[r * 32 + 16 + wp] - Tb[(32 + r) * 32 + wp];
  int m = r * 16 + wp;
  long dst = ((long)m * 16 + b) * 64 + c;
  ws[O_U2RE + dst] = re;
  ws[O_U2IM + dst] = im;
}

// ---------------------------------------------------------------------------
// Stage 3: per mode m: V(16b x 64o) = U2_m(16 x 64) x Wt_m(64 x 64), complex.
// 512 waves; 16 WMMA per k-step. A-negate done in VALU (f32 WMMA has CNeg only).
// ---------------------------------------------------------------------------
__global__ void k_spectral(float* __restrict__ ws) {
  int gid  = blockIdx.x * blockDim.x + threadIdx.x;
  int wave = gid >> 5;
  int lane = threadIdx.x & 31;
  int ln = lane & 15, hi = lane >> 4;
  int m = wave;
  const float* Ure = ws + O_U2RE + (long)m * 16 * 64;
  const float* Uim = ws + O_U2IM + (long)m * 16 * 64;
  const float* Bre = ws + O_WTRE + (long)m * 64 * 64;
  const float* Bim = ws + O_WTIM + (long)m * 64 * 64;
  v8f vre[4] = {v8f{}, v8f{}, v8f{}, v8f{}};
  v8f vim[4] = {v8f{}, v8f{}, v8f{}, v8f{}};
  for (int k0 = 0; k0 < 64; k0 += 4) {
    int k = k0 + hi * 2;
    v2f are  = *(const v2f*)(Ure + ln * 64 + k);
    v2f aim  = *(const v2f*)(Uim + ln * 64 + k);
    v2f aimn; aimn.x = -aim.x; aimn.y = -aim.y;
#pragma unroll
    for (int nt = 0; nt < 4; ++nt) {
      int n = nt * 16 + ln;
      v2f br, bi;
      br.x = Bre[k * 64 + n];  br.y = Bre[(k + 1) * 64 + n];
      bi.x = Bim[k * 64 + n];  bi.y = Bim[(k + 1) * 64 + n];
      vre[nt] = wmma_f32(are,  br, vre[nt]);
      vre[nt] = wmma_f32(aimn, bi, vre[nt]);   // re -= Uim*Wim
      vim[nt] = wmma_f32(are,  bi, vim[nt]);
      vim[nt] = wmma_f32(aim,  br, vim[nt]);
    }
  }
  int r = m >> 4, wp = m & 15;
#pragma unroll
  for (int nt = 0; nt < 4; ++nt)
#pragma unroll
    for (int v = 0; v < 8; ++v) {
      int b = v + 8 * hi;
      int o = nt * 16 + ln;
      long dst = (((long)b * 32 + r) * 16 + wp) * 64 + o;
      ws[O_VRE + dst] = vre[nt][v];
      ws[O_VIM + dst] = vim[nt][v];
    }
}

// ---------------------------------------------------------------------------
// Stage 4: per (b,o), per half (re/im): G_half(256x16) = E(256x64) x [Vre;Vim](64x16).
// 32768 waves. G reuses the S1 buffer, layout [bo][x][re0..15|im0..15].
// ---------------------------------------------------------------------------
__global__ void k_idft_h(float* __restrict__ ws) {
  int gid  = blockIdx.x * blockDim.x + threadIdx.x;
  int wave = gid >> 5;
  int lane = threadIdx.x & 31;
  int ln = lane & 15, hi = lane >> 4;
  int mt = wave & 15, half = (wave >> 4) & 1, bo = wave >> 5;
  const float* coef = ws + (half ? O_EIM : O_ERE);
  const float* Vr = ws + O_VRE;
  const float* Vi = ws + O_VIM;
  int b = bo >> 6, o = bo & 63;
  long vbase = (long)b * 32768 + o;
  v8f acc = {};
  for (int k0 = 0; k0 < 64; k0 += 4) {
    int k = k0 + hi * 2;                      // pairs never straddle the 31/32 boundary
    v2f a = *(const v2f*)(coef + (mt * 16 + ln) * 64 + k);
    const float* Vc = (k < 32) ? Vr : Vi;
    int r = k & 31;
    v2f bb;
    bb.x = Vc[vbase + (long)r * 512 + ln * 64];
    bb.y = Vc[vbase + (long)(r + 1) * 512 + ln * 64];
    acc = wmma_f32(a, bb, acc);
  }
  float* G = ws + O_S1;
  long gbase = ((long)bo * 256 + mt * 16) * 32;
#pragma unroll
  for (int v = 0; v < 8; ++v) {
    int x = v + 8 * hi;
    G[gbase + (long)x * 32 + half * 16 + ln] = acc[v];
  }
}

// ---------------------------------------------------------------------------
// Stage 5: out(M1 x 256) = G(M1 x 32) x Fw(32 x 256). 65536 waves;
// 4 waves (one block) share an M-tile, each covers 64 output columns.
// ---------------------------------------------------------------------------
__global__ void k_idft_w(float* __restrict__ ws, float* __restrict__ out) {
  const float* G  = ws + O_S1;
  const float* Fw = ws + O_FW;
  int gid  = blockIdx.x * blockDim.x + threadIdx.x;
  int wave = gid >> 5;
  int lane = threadIdx.x & 31;
  int ln = lane & 15, hi = lane >> 4;
  int ng = wave & 3;
  long mt = wave >> 2;
  long rowbase = mt * 16;
  const float* arow = G + (rowbase + ln) * 32;
  v8f acc[4] = {v8f{}, v8f{}, v8f{}, v8f{}};
  for (int k0 = 0; k0 < 32; k0 += 4) {
    int k = k0 + hi * 2;
    v2f a = *(const v2f*)(arow + k);
#pragma unroll
    for (int nt = 0; nt < 4; ++nt) {
      int n = ng * 64 + nt * 16 + ln;
      v2f bb;
      bb.x = Fw[k * 256 + n];
      bb.y = Fw[(k + 1) * 256 + n];
      acc[nt] = wmma_f32(a, bb, acc[nt]);
    }
  }
#pragma unroll
  for (int nt = 0; nt < 4; ++nt)
#pragma unroll
    for (int v = 0; v < 8; ++v) {
      long row = rowbase + v + 8 * hi;
      out[row * 256 + ng * 64 + nt * 16 + ln] = acc[nt][v];
    }
}

// ---------------------------------------------------------------------------
extern "C" void kernel_launch(void* const* d_in, const int* in_sizes, int n_in,
                              void* d_out, int out_size, void* d_ws, size_t ws_size,
                              hipStream_t stream) {
  (void)in_sizes; (void)n_in; (void)out_size; (void)ws_size;
  const float* u   = (const float*)d_in[0];
  const float* Wpr = (const float*)d_in[1];
  const float* Wpi = (const float*)d_in[2];
  const float* Wnr = (const float*)d_in[3];
  const float* Wni = (const float*)d_in[4];
  float* out = (float*)d_out;
  float* ws  = (float*)d_ws;

  k_prep_tables <<<256,   256, 0, stream>>>(ws);
  k_prep_weights<<<8192,  256, 0, stream>>>(Wpr, Wpi, Wnr, Wni, ws);
  k_dft_w       <<<4096,  128, 0, stream>>>(u, ws);     // 16384 waves
  k_dft_h       <<<1024,  128, 0, stream>>>(ws);        //  4096 waves
  k_combine     <<<2048,  256, 0, stream>>>(ws);
  k_spectral    <<<128,   128, 0, stream>>>(ws);        //   512 waves
  k_idft_h      <<<8192,  128, 0, stream>>>(ws);        // 32768 waves
  k_idft_w      <<<16384, 128, 0, stream>>>(ws, out);   // 65536 waves
}